// DynamicMoERouter_17248588661239
// MI455X (gfx1250) — compile-verified
//
#include <hip/hip_runtime.h>
#include <hip/hip_bf16.h>

typedef __attribute__((ext_vector_type(2))) float v2f;
typedef __attribute__((ext_vector_type(8))) float v8f;

#define D_DIM 2048
#define E_DIM 64

// ---------------------------------------------------------------------------
// Kernel 1: repack W [64][2048] row-major -> Wt2 laid out as [k/2][e][2]
// so that a B-fragment lane load (k, k+1) for expert n is one contiguous b64.
// ---------------------------------------------------------------------------
__global__ __launch_bounds__(256) void moe_transpose_w(const float* __restrict__ W,
                                                       float* __restrict__ Wt2) {
    int idx = blockIdx.x * 256 + threadIdx.x;          // 0 .. 64*2048-1
    if (idx < E_DIM * D_DIM) {
        int e = idx >> 11;       // / 2048
        int k = idx & (D_DIM - 1);
        Wt2[((k >> 1) * E_DIM + e) * 2 + (k & 1)] = W[idx];
    }
}

// ---------------------------------------------------------------------------
// Kernel 2: logits = x @ W^T via V_WMMA_F32_16X16X4_F32.
// Block = 256 threads = 8 waves; each wave computes 32 rows x all 64 experts
// (2 row-groups x 4 expert tiles = 8 C tiles, 64 acc VGPRs). Each B fragment
// is reused across both row groups, halving L2 B traffic vs 16 rows/wave.
// A frag (16x4 f32): lane l holds row m=l&15, K = k0 + 2*(l>>4) + {0,1}
// B frag (4x16 f32): lane l holds col n=l&15, same K mapping.
// C/D (16x16 f32, 8 VGPRs): row = vgpr + 8*(l>>4), col = l&15.
// ---------------------------------------------------------------------------
__global__ __launch_bounds__(256) void moe_gemm_wmma(const float* __restrict__ x,
                                                     const float* __restrict__ Wt2,
                                                     float* __restrict__ logits) {
    const int lane = threadIdx.x & 31;
    const int wave = threadIdx.x >> 5;
    const int rowBase = blockIdx.x * 256 + wave * 32;
    const int m  = lane & 15;
    const int kp = lane >> 4;

    const float* xa0 = x + (size_t)(rowBase + m) * D_DIM + 2 * kp;        // rows 0..15
    const float* xa1 = xa0 + (size_t)16 * D_DIM;                          // rows 16..31
    // Per-tile B base (floats) in Wt2: ((p*64 + n)*2), p = k0/2 + kp -> offset k0*64
    const float* wb0 = Wt2 + ((size_t)kp * E_DIM + 0 * 16 + m) * 2;
    const float* wb1 = Wt2 + ((size_t)kp * E_DIM + 1 * 16 + m) * 2;
    const float* wb2 = Wt2 + ((size_t)kp * E_DIM + 2 * 16 + m) * 2;
    const float* wb3 = Wt2 + ((size_t)kp * E_DIM + 3 * 16 + m) * 2;

    v8f a0t0 = {}, a0t1 = {}, a0t2 = {}, a0t3 = {};   // row group 0, tiles 0..3
    v8f a1t0 = {}, a1t1 = {}, a1t2 = {}, a1t3 = {};   // row group 1, tiles 0..3

#pragma unroll 4
    for (int k0 = 0; k0 < D_DIM; k0 += 4) {
        v2f a0 = *(const v2f*)(xa0 + k0);
        v2f a1 = *(const v2f*)(xa1 + k0);
        const size_t wo = (size_t)k0 * E_DIM;
        v2f b0 = *(const v2f*)(wb0 + wo);
        v2f b1 = *(const v2f*)(wb1 + wo);
        v2f b2 = *(const v2f*)(wb2 + wo);
        v2f b3 = *(const v2f*)(wb3 + wo);
        a0t0 = __builtin_amdgcn_wmma_f32_16x16x4_f32(false, a0, false, b0, (short)0, a0t0, false, false);
        a1t0 = __builtin_amdgcn_wmma_f32_16x16x4_f32(false, a1, false, b0, (short)0, a1t0, false, false);
        a0t1 = __builtin_amdgcn_wmma_f32_16x16x4_f32(false, a0, false, b1, (short)0, a0t1, false, false);
        a1t1 = __builtin_amdgcn_wmma_f32_16x16x4_f32(false, a1, false, b1, (short)0, a1t1, false, false);
        a0t2 = __builtin_amdgcn_wmma_f32_16x16x4_f32(false, a0, false, b2, (short)0, a0t2, false, false);
        a1t2 = __builtin_amdgcn_wmma_f32_16x16x4_f32(false, a1, false, b2, (short)0, a1t2, false, false);
        a0t3 = __builtin_amdgcn_wmma_f32_16x16x4_f32(false, a0, false, b3, (short)0, a0t3, false, false);
        a1t3 = __builtin_amdgcn_wmma_f32_16x16x4_f32(false, a1, false, b3, (short)0, a1t3, false, false);
    }

    // Scatter C tiles to logits [N][64] (raw, bias added in finalize kernel)
    const int rBase = rowBase + 8 * kp;
#pragma unroll
    for (int v = 0; v < 8; ++v) {
        float* rowp0 = logits + (size_t)(rBase + v) * E_DIM + m;
        rowp0[0]  = a0t0[v];
        rowp0[16] = a0t1[v];
        rowp0[32] = a0t2[v];
        rowp0[48] = a0t3[v];
        float* rowp1 = rowp0 + (size_t)16 * E_DIM;
        rowp1[0]  = a1t0[v];
        rowp1[16] = a1t1[v];
        rowp1[32] = a1t2[v];
        rowp1[48] = a1t3[v];
    }
}

// ---------------------------------------------------------------------------
// Kernel 3: per-row softmax + top-2 (lowest-index tie-break, like lax.top_k).
// One wave per row; lane owns experts (lane) and (lane+32).
// Reads raw logits from the all_probs region, writes probs back in place.
// ---------------------------------------------------------------------------
struct Top2 { float v1; int i1; float v2; int i2; };

__device__ __forceinline__ bool better(float v, int i, float ov, int oi) {
    return (v > ov) || (v == ov && i < oi);
}

__device__ __forceinline__ Top2 merge2(Top2 a, Top2 b) {
    Top2 r;
    if (better(a.v1, a.i1, b.v1, b.i1)) {
        r.v1 = a.v1; r.i1 = a.i1;
        if (better(a.v2, a.i2, b.v1, b.i1)) { r.v2 = a.v2; r.i2 = a.i2; }
        else                                { r.v2 = b.v1; r.i2 = b.i1; }
    } else {
        r.v1 = b.v1; r.i1 = b.i1;
        if (better(b.v2, b.i2, a.v1, a.i1)) { r.v2 = b.v2; r.i2 = b.i2; }
        else                                { r.v2 = a.v1; r.i2 = a.i1; }
    }
    return r;
}

__global__ __launch_bounds__(256) void moe_finalize(const float* __restrict__ bias,
                                                    float* __restrict__ rw,
                                                    int* __restrict__ topk,
                                                    float* __restrict__ probs) {
    const int lane = threadIdx.x & 31;
    const int wave = threadIdx.x >> 5;
    const int row  = blockIdx.x * 8 + wave;

    float* rowp = probs + (size_t)row * E_DIM;
    const float v0 = rowp[lane]      + bias[lane];
    const float v1 = rowp[lane + 32] + bias[lane + 32];

    // --- full softmax over 64 experts ---
    float mx = fmaxf(v0, v1);
#pragma unroll
    for (int s = 16; s > 0; s >>= 1) mx = fmaxf(mx, __shfl_xor(mx, s, 32));
    const float e0 = __builtin_expf(v0 - mx);
    const float e1 = __builtin_expf(v1 - mx);
    float sum = e0 + e1;
#pragma unroll
    for (int s = 16; s > 0; s >>= 1) sum += __shfl_xor(sum, s, 32);
    const float inv = 1.0f / sum;
    rowp[lane]      = e0 * inv;
    rowp[lane + 32] = e1 * inv;

    // --- top-2 reduction ---
    Top2 r;
    if (better(v0, lane, v1, lane + 32)) { r.v1 = v0; r.i1 = lane;      r.v2 = v1; r.i2 = lane + 32; }
    else                                 { r.v1 = v1; r.i1 = lane + 32; r.v2 = v0; r.i2 = lane; }
#pragma unroll
    for (int s = 16; s > 0; s >>= 1) {
        Top2 o;
        o.v1 = __shfl_xor(r.v1, s, 32);
        o.i1 = __shfl_xor(r.i1, s, 32);
        o.v2 = __shfl_xor(r.v2, s, 32);
        o.i2 = __shfl_xor(r.i2, s, 32);
        r = merge2(r, o);
    }

    // --- softmax over the two selected logits ---
    const float el = __builtin_expf(r.v2 - r.v1);
    const float p1 = 1.0f / (1.0f + el);
    const float p2 = el / (1.0f + el);

    float* rwp = rw + (size_t)row * E_DIM;
    const int eA = lane, eB = lane + 32;
    rwp[eA] = (eA == r.i1) ? p1 : ((eA == r.i2) ? p2 : 0.0f);
    rwp[eB] = (eB == r.i1) ? p1 : ((eB == r.i2) ? p2 : 0.0f);

    if (lane == 0) {
        topk[row * 2 + 0] = r.i1;
        topk[row * 2 + 1] = r.i2;
    }
}

// ---------------------------------------------------------------------------
extern "C" void kernel_launch(void* const* d_in, const int* in_sizes, int n_in,
                              void* d_out, int out_size, void* d_ws, size_t ws_size,
                              hipStream_t stream) {
    const float* x = (const float*)d_in[0];   // [N, 2048]
    const float* W = (const float*)d_in[1];   // [64, 2048]
    const float* b = (const float*)d_in[2];   // [64]
    const int N = in_sizes[0] / D_DIM;        // 32768

    float* out   = (float*)d_out;
    float* rw    = out;                              // [N,64] f32
    int*   topk  = (int*)(out + (size_t)N * E_DIM);  // [N,2] i32
    float* probs = out + (size_t)N * E_DIM + (size_t)N * 2; // [N,64] f32 (logits -> probs)

    float* Wt2 = (float*)d_ws;                       // 2048*64*4 = 512 KB

    moe_transpose_w<<<(E_DIM * D_DIM + 255) / 256, 256, 0, stream>>>(W, Wt2);
    moe_gemm_wmma<<<N / 256, 256, 0, stream>>>(x, Wt2, probs);
    moe_finalize<<<N / 8, 256, 0, stream>>>(b, rw, topk, probs);
}